// BehlerG1_62328565399849
// MI455X (gfx1250) — compile-verified
//
#include <hip/hip_runtime.h>
#include <math.h>

typedef __attribute__((ext_vector_type(2))) float v2f;
typedef __attribute__((ext_vector_type(8))) float v8f;

#define RC_CUT 5.0f
#define NB 64      // neighbors per atom
#define NR 64      // radial functions
#define NA 1024    // atoms per batch
#define WAVES 8    // waves (atoms) per block

__global__ __launch_bounds__(256)
void behler_g1_kernel(const float* __restrict__ positions,      // (B,A,3)
                      const float* __restrict__ cell,           // (B,3,3)
                      const float* __restrict__ offsets,        // (B,A,N,3)
                      const float* __restrict__ mask,           // (B,A,N)
                      const float* __restrict__ etas,           // (R)
                      const float* __restrict__ rss,            // (R)
                      const float* __restrict__ z_emb,          // (300,1)
                      const long long* __restrict__ neighbors,  // (B,A,N) i64
                      const long long* __restrict__ atomic_numbers, // (B,A) i64
                      float* __restrict__ out)                  // (B,A,R)
{
    // A operands staged directly in V_WMMA_F32_16X16X4_F32 A-matrix layout:
    //   tile t, lane m   (m<16): {A[m,0], A[m,1]} = {off.x, off.y}
    //   tile t, lane m+16      : {A[m,2], A[m,3]} = {off.z, 0}
    __shared__ __align__(16) float sA[WAVES][4][32][2];
    __shared__ __align__(16) float sShift[WAVES][NB][4]; // shift = offsets @ cell
    __shared__ float sD[WAVES][NB];                      // masked distances
    __shared__ float sW[WAVES][NB];                      // cutoff * mask * z_j

    const int tid  = threadIdx.x;
    const int wave = tid >> 5;
    const int lane = tid & 31;

    const int atomGlobal = blockIdx.x * WAVES + wave;
    const int b = atomGlobal / NA;
    const int a = atomGlobal % NA;

    // ---------------- Phase 1: neighbor gathers (2 neighbors / lane) --------
    const float pix = positions[(b*NA + a)*3 + 0];
    const float piy = positions[(b*NA + a)*3 + 1];
    const float piz = positions[(b*NA + a)*3 + 2];

    float dpx[2], dpy[2], dpz[2], wz[2], mk[2];
    #pragma unroll
    for (int k = 0; k < 2; ++k) {
        const int n = lane + 32*k;
        const int t = n >> 4;          // WMMA tile
        const int m = n & 15;          // row within tile
        const long long pairBase = ((long long)(b*NA + a))*NB + n;
        const int j  = (int)neighbors[pairBase];
        dpx[k] = positions[(b*NA + j)*3 + 0] - pix;
        dpy[k] = positions[(b*NA + j)*3 + 1] - piy;
        dpz[k] = positions[(b*NA + j)*3 + 2] - piz;
        mk[k]  = mask[pairBase];
        const int zn = (int)atomic_numbers[b*NA + j];
        wz[k]  = mk[k] * z_emb[zn];          // z_emb has E==1
        const float* op = &offsets[pairBase*3];
        sA[wave][t][m][0]      = op[0];      // K=0
        sA[wave][t][m][1]      = op[1];      // K=1
        sA[wave][t][m+16][0]   = op[2];      // K=2
        sA[wave][t][m+16][1]   = 0.0f;       // K=3 padding
    }
    __syncthreads();

    // ------- Phase 2: shift = offsets @ cell via V_WMMA_F32_16X16X4_F32 ----
    // B matrix (4x16), exact f32: VGPR0 = row0 (lanes 0-15) / row2 (16-31),
    //                             VGPR1 = row1 (lanes 0-15) / row3=0 (16-31)
    const int col  = lane & 15;
    const int half = lane >> 4;
    v2f Bm;
    {
        float b0 = 0.0f, b1 = 0.0f;
        if (col < 3) {
            const float* cb = &cell[b*9];
            b0 = half ? cb[2*3 + col] : cb[0*3 + col];
            b1 = half ? 0.0f          : cb[1*3 + col];
        }
        Bm.x = b0; Bm.y = b1;
    }
    #pragma unroll
    for (int t = 0; t < 4; ++t) {
        const v2f Am = *(const v2f*)&sA[wave][t][lane][0];   // branchless b64 load
        v8f Cm = {0.f,0.f,0.f,0.f,0.f,0.f,0.f,0.f};
        v8f Dm = __builtin_amdgcn_wmma_f32_16x16x4_f32(
                     false, Am, false, Bm, (short)0, Cm, false, false);
        // D (16x16): VGPR v -> row (v + 8*half), column = lane&15
        if (col < 3) {
            #pragma unroll
            for (int v = 0; v < 8; ++v)
                sShift[wave][16*t + v + 8*half][col] = Dm[v];
        }
    }
    __syncthreads();

    // ---------------- Phase 3: distances, cutoff, weights ------------------
    #pragma unroll
    for (int k = 0; k < 2; ++k) {
        const int n = lane + 32*k;
        const float vx = dpx[k] + sShift[wave][n][0];
        const float vy = dpy[k] + sShift[wave][n][1];
        const float vz = dpz[k] + sShift[wave][n][2];
        const float d2 = vx*vx + vy*vy + vz*vz;
        // distances = sqrt(where(mask>0.5, d2, 1.0)) * mask
        const float d  = ((mk[k] > 0.5f) ? sqrtf(d2) : 1.0f) * mk[k];
        const float cut = (d < RC_CUT)
            ? 0.5f * (__cosf(d * (3.14159265358979f / RC_CUT)) + 1.0f)
            : 0.0f;
        sD[wave][n] = d;
        sW[wave][n] = cut * wz[k];   // cutoff * mask * z_j
    }
    __syncthreads();

    // --------- Phase 4: radial Gaussians + reduction over neighbors --------
    const float eta0 = etas[lane],      rs0 = rss[lane];
    const float eta1 = etas[lane + 32], rs1 = rss[lane + 32];
    float acc0 = 0.0f, acc1 = 0.0f;
    #pragma unroll 4
    for (int n = 0; n < NB; ++n) {
        const float d = sD[wave][n];
        const float w = sW[wave][n];
        const float t0 = d - rs0;
        const float t1 = d - rs1;
        acc0 += __expf(-eta0 * t0 * t0) * w;
        acc1 += __expf(-eta1 * t1 * t1) * w;
    }
    float* op = &out[((long long)(b*NA + a))*NR];
    op[lane]      = acc0;
    op[lane + 32] = acc1;
}

extern "C" void kernel_launch(void* const* d_in, const int* in_sizes, int n_in,
                              void* d_out, int out_size, void* d_ws, size_t ws_size,
                              hipStream_t stream) {
    const float*     positions      = (const float*)d_in[0];
    const float*     cell           = (const float*)d_in[1];
    const float*     offsets        = (const float*)d_in[2];
    const float*     maskp          = (const float*)d_in[3];
    const float*     etas           = (const float*)d_in[4];
    const float*     rss            = (const float*)d_in[5];
    const float*     z_emb          = (const float*)d_in[6];
    const long long* neighbors      = (const long long*)d_in[7];
    const long long* atomic_numbers = (const long long*)d_in[8];
    float* out = (float*)d_out;

    const int B = 16;
    const int totalAtoms = B * NA;                 // 16384 waves, 1 atom each
    dim3 grid(totalAtoms / WAVES);                 // 2048 blocks
    dim3 block(256);                               // 8 wave32 waves
    behler_g1_kernel<<<grid, block, 0, stream>>>(
        positions, cell, offsets, maskp, etas, rss, z_emb,
        neighbors, atomic_numbers, out);
}